// ProbSparseSelfAttention_34462817583505
// MI455X (gfx1250) — compile-verified
//
#include <hip/hip_runtime.h>

typedef __attribute__((ext_vector_type(2))) float v2f;
typedef __attribute__((ext_vector_type(8))) float v8f;
typedef __attribute__((ext_vector_type(4))) int v4i;

#define B_  4
#define H_  8
#define L_  2048
#define F_  64
#define NS_ 40      // sampled keys per query
#define NQ_ 40      // top queries per (b,h)
#define D_  512
#define BH_ (B_*H_)
#define NCH_ 8      // key-dim chunks for attention
#define CHK_ (L_/NCH_)   // 256 keys per chunk
#define PAD_ 68     // LDS row stride in floats: 272B, 16B-aligned, bank stride 4 -> conflict free

// fp32 WMMA: D(16x16,f32) = A(16x4,f32) * B(4x16,f32) + C
__device__ __forceinline__ v8f wmma4(v2f a, v2f b, v8f c) {
  return __builtin_amdgcn_wmma_f32_16x16x4_f32(false, a, false, b, (short)0, c, false, false);
}

// ---- async global->LDS 16B copy (CDNA5 GLOBAL_LOAD_ASYNC_TO_LDS_B128), with fallback ----
#if defined(__HIP_DEVICE_COMPILE__) && __has_builtin(__builtin_amdgcn_global_load_async_to_lds_b128)
#define HAS_ASYNC 1
#else
#define HAS_ASYNC 0
#endif

#if HAS_ASYNC
typedef __attribute__((address_space(1))) v4i* gv4i_p;
typedef __attribute__((address_space(3))) v4i* lv4i_p;
#endif

__device__ __forceinline__ void cp16(void* lds_dst, const void* gsrc) {
#if HAS_ASYNC
  __builtin_amdgcn_global_load_async_to_lds_b128((gv4i_p)(gsrc), (lv4i_p)(lds_dst), 0, 0);
#else
  *(float4*)lds_dst = *(const float4*)gsrc;
#endif
}

__device__ __forceinline__ void cp_wait() {
#if HAS_ASYNC
#if __has_builtin(__builtin_amdgcn_s_wait_asynccnt)
  __builtin_amdgcn_s_wait_asynccnt(0);
#else
  asm volatile("s_wait_asynccnt 0" ::: "memory");
#endif
#endif
}

// ---------------- 1: per-(b,h) mean of v over L ----------------
__global__ void vmean_kernel(const float* __restrict__ v, float* __restrict__ vmean) {
  __shared__ float red[256];
  int bh = blockIdx.x, tid = threadIdx.x;
  int f = tid & 63, seg = tid >> 6;
  const float* vb = v + (size_t)bh * L_ * F_;
  float s = 0.f;
  for (int l = seg; l < L_; l += 4) s += vb[l * F_ + f];
  red[tid] = s;
  __syncthreads();
  if (tid < 64)
    vmean[bh * 64 + tid] =
        (red[tid] + red[tid + 64] + red[tid + 128] + red[tid + 192]) * (1.0f / L_);
}

// ---------------- 2: fill ctx with broadcast v_mean (float4 stores) ----------------
__global__ void ctxfill_kernel(const float* __restrict__ vmean, float4* __restrict__ ctx4) {
  int g4 = blockIdx.x * 256 + threadIdx.x;   // over BH*L*F/4 = 1M float4
  int g = g4 * 4;
  ctx4[g4] = *(const float4*)&vmean[(((g >> 17)) << 6) | (g & 63)];   // L*F = 2^17
}

// ---------------- 3: sparsity measure m = max_s qk - sum_s qk / L ----------------
__global__ void sparsity_kernel(const float* __restrict__ q, const float* __restrict__ k,
                                const int* __restrict__ key_idx, float* __restrict__ mbuf) {
  int wave = threadIdx.x >> 5, lane = threadIdx.x & 31;
  int g  = blockIdx.x * 8 + wave;        // global query index over BH*L
  int bh = g >> 11, l = g & (L_ - 1);
  const float* qrow = q + (size_t)g * F_;
  float q0 = qrow[2 * lane], q1 = qrow[2 * lane + 1];
  const float* kb = k + (size_t)bh * L_ * F_;
  const int* idxrow = key_idx + l * NS_;
  float mx = -3.4e38f, sm = 0.f;
  for (int s = 0; s < NS_; ++s) {
    const float* krow = kb + (size_t)idxrow[s] * F_;
    float p = q0 * krow[2 * lane] + q1 * krow[2 * lane + 1];
#pragma unroll
    for (int off = 16; off > 0; off >>= 1) p += __shfl_xor(p, off, 32);
    mx = fmaxf(mx, p);
    sm += p;
  }
  if (lane == 0) mbuf[g] = mx - sm * (1.0f / L_);
}

// ---------------- 4: top-40 per (b,h), iterative argmax ----------------
__global__ void topk_kernel(const float* __restrict__ mbuf, int* __restrict__ topIdx) {
  __shared__ float sm[L_];
  __shared__ float sv[256];
  __shared__ int   si[256];
  int bh = blockIdx.x, tid = threadIdx.x;
  for (int i = tid; i < L_; i += 256) sm[i] = mbuf[bh * L_ + i];
  __syncthreads();
  for (int it = 0; it < NQ_; ++it) {
    float best = -3.4e38f; int bi = 0;
    for (int i = tid; i < L_; i += 256)
      if (sm[i] > best) { best = sm[i]; bi = i; }
    sv[tid] = best; si[tid] = bi;
    __syncthreads();
    for (int s = 128; s > 0; s >>= 1) {
      if (tid < s && sv[tid + s] > sv[tid]) { sv[tid] = sv[tid + s]; si[tid] = si[tid + s]; }
      __syncthreads();
    }
    if (tid == 0) { topIdx[bh * NQ_ + it] = si[0]; sm[si[0]] = -3.4e38f; }
    __syncthreads();
  }
}

// ---- shared helper: load scaled q_top (48 padded rows) into LDS ----
__device__ __forceinline__ void load_qtop(const float* __restrict__ q, const int* __restrict__ ti,
                                          int bh, int tid, float* Qs) {
  for (int i = tid; i < 48 * 64; i += 96) {
    int m = i >> 6, f = i & 63;
    float val = 0.f;
    if (m < NQ_) val = q[((size_t)bh * L_ + ti[m]) * F_ + f] * 0.125f;  // 1/sqrt(64)
    Qs[m * PAD_ + f] = val;
  }
}

// ---------------- 5a: per-chunk row max of scores ----------------
__global__ void attn_max_kernel(const float* __restrict__ q, const float* __restrict__ k,
                                const int* __restrict__ topIdx, float* __restrict__ pmax) {
  __shared__ float Qs[48 * PAD_];
  __shared__ float Ks[64 * PAD_];
  int bh = blockIdx.x, ch = blockIdx.y, tid = threadIdx.x;
  int wave = tid >> 5, lane = tid & 31;
  int half = lane >> 4, mn = lane & 15, koff = half * 2;
  const float* kb = k + (size_t)bh * L_ * F_;
  const int* ti = topIdx + bh * NQ_;
  load_qtop(q, ti, bh, tid, Qs);

  float rmax[8];
#pragma unroll
  for (int i = 0; i < 8; ++i) rmax[i] = -3.4e38f;
  for (int n0 = ch * CHK_; n0 < (ch + 1) * CHK_; n0 += 64) {
    __syncthreads();
    for (int i = tid; i < 256; i += 96) {   // 64 rows x 4 float4
      int r = i >> 2, c4 = (i & 3) * 4;
      cp16(&Ks[r * PAD_ + c4], &kb[(n0 + r) * F_ + c4]);
    }
    cp_wait();
    __syncthreads();
    for (int nt = 0; nt < 4; ++nt) {
      v8f s0 = {}, s1 = {};
#pragma unroll
      for (int kk = 0; kk < 64; kk += 8) {
        v2f a, b;
        a.x = Qs[(wave * 16 + mn) * PAD_ + kk + koff];
        a.y = Qs[(wave * 16 + mn) * PAD_ + kk + koff + 1];
        b.x = Ks[(nt * 16 + mn) * PAD_ + kk + koff];
        b.y = Ks[(nt * 16 + mn) * PAD_ + kk + koff + 1];
        s0 = wmma4(a, b, s0);
        a.x = Qs[(wave * 16 + mn) * PAD_ + kk + 4 + koff];
        a.y = Qs[(wave * 16 + mn) * PAD_ + kk + 4 + koff + 1];
        b.x = Ks[(nt * 16 + mn) * PAD_ + kk + 4 + koff];
        b.y = Ks[(nt * 16 + mn) * PAD_ + kk + 4 + koff + 1];
        s1 = wmma4(a, b, s1);
      }
      v8f s = s0 + s1;
#pragma unroll
      for (int i = 0; i < 8; ++i) {
        float t = s[i];
        t = fmaxf(t, __shfl_xor(t, 1, 32));
        t = fmaxf(t, __shfl_xor(t, 2, 32));
        t = fmaxf(t, __shfl_xor(t, 4, 32));
        t = fmaxf(t, __shfl_xor(t, 8, 32));
        rmax[i] = fmaxf(rmax[i], t);
      }
    }
  }
#pragma unroll
  for (int i = 0; i < 8; ++i)
    if (mn == 0) pmax[(bh * NCH_ + ch) * 48 + wave * 16 + i + 8 * half] = rmax[i];
}

// ---------------- 5b: per-chunk exp / P@V partials; row-sum folded into WMMA ----------------
__global__ void attn_pv_kernel(const float* __restrict__ q, const float* __restrict__ k,
                               const float* __restrict__ v, const int* __restrict__ topIdx,
                               const float* __restrict__ pmax,
                               float* __restrict__ psum, float* __restrict__ pO) {
  __shared__ float Qs[48 * PAD_];
  __shared__ float Ks[64 * PAD_];
  __shared__ float Vs[64 * PAD_];
  __shared__ float Ps[3][16 * 17];
  __shared__ float gmax48[48];
  int bh = blockIdx.x, ch = blockIdx.y, tid = threadIdx.x;
  int wave = tid >> 5, lane = tid & 31;
  int half = lane >> 4, mn = lane & 15, koff = half * 2;
  const float* kb = k + (size_t)bh * L_ * F_;
  const float* vb = v + (size_t)bh * L_ * F_;
  const int* ti = topIdx + bh * NQ_;
  load_qtop(q, ti, bh, tid, Qs);
  if (tid < 48) {
    float m = -3.4e38f;
    for (int c = 0; c < NCH_; ++c) m = fmaxf(m, pmax[(bh * NCH_ + c) * 48 + tid]);
    gmax48[tid] = m;
  }
  __syncthreads();
  float rmax[8];
#pragma unroll
  for (int i = 0; i < 8; ++i) rmax[i] = gmax48[wave * 16 + i + 8 * half];

  v8f o0 = {}, o1 = {}, o2 = {}, o3 = {}, o4 = {};   // o4: row-sum accumulator (B = ones)
  const v2f bones = {1.0f, 1.0f};
  float* Psw = Ps[wave];
  for (int n0 = ch * CHK_; n0 < (ch + 1) * CHK_; n0 += 64) {
    __syncthreads();
    for (int i = tid; i < 512; i += 96) {   // Ks + Vs: 2 x 64 rows x 4 float4
      int r = (i >> 2) & 63, c4 = (i & 3) * 4;
      if (i < 256) cp16(&Ks[r * PAD_ + c4], &kb[(n0 + r) * F_ + c4]);
      else         cp16(&Vs[r * PAD_ + c4], &vb[(n0 + r) * F_ + c4]);
    }
    cp_wait();
    __syncthreads();
    for (int nt = 0; nt < 4; ++nt) {
      v8f s0 = {}, s1 = {};
#pragma unroll
      for (int kk = 0; kk < 64; kk += 8) {
        v2f a, b;
        a.x = Qs[(wave * 16 + mn) * PAD_ + kk + koff];
        a.y = Qs[(wave * 16 + mn) * PAD_ + kk + koff + 1];
        b.x = Ks[(nt * 16 + mn) * PAD_ + kk + koff];
        b.y = Ks[(nt * 16 + mn) * PAD_ + kk + koff + 1];
        s0 = wmma4(a, b, s0);
        a.x = Qs[(wave * 16 + mn) * PAD_ + kk + 4 + koff];
        a.y = Qs[(wave * 16 + mn) * PAD_ + kk + 4 + koff + 1];
        b.x = Ks[(nt * 16 + mn) * PAD_ + kk + 4 + koff];
        b.y = Ks[(nt * 16 + mn) * PAD_ + kk + 4 + koff + 1];
        s1 = wmma4(a, b, s1);
      }
      v8f s = s0 + s1;
#pragma unroll
      for (int i = 0; i < 8; ++i)
        Psw[(i + 8 * half) * 17 + mn] = __expf(s[i] - rmax[i]);   // (m, key) order
      __syncthreads();
#pragma unroll
      for (int kk = 0; kk < 16; kk += 4) {
        v2f a;
        a.x = Psw[mn * 17 + kk + koff];
        a.y = Psw[mn * 17 + kk + koff + 1];
        int kr0 = (nt * 16 + kk + koff) * PAD_;
        int kr1 = kr0 + PAD_;
        v2f b0, b1, b2, b3;
        b0.x = Vs[kr0 +  0 + mn]; b0.y = Vs[kr1 +  0 + mn];
        b1.x = Vs[kr0 + 16 + mn]; b1.y = Vs[kr1 + 16 + mn];
        b2.x = Vs[kr0 + 32 + mn]; b2.y = Vs[kr1 + 32 + mn];
        b3.x = Vs[kr0 + 48 + mn]; b3.y = Vs[kr1 + 48 + mn];
        o0 = wmma4(a, b0, o0);
        o1 = wmma4(a, b1, o1);
        o2 = wmma4(a, b2, o2);
        o3 = wmma4(a, b3, o3);
        o4 = wmma4(a, bones, o4);   // every column of o4 = running row-sum of P
      }
      __syncthreads();
    }
  }
#pragma unroll
  for (int i = 0; i < 8; ++i) {
    int qrow = wave * 16 + i + 8 * half;
    int prow = (bh * NCH_ + ch) * 48 + qrow;
    if (mn == 0) psum[prow] = o4[i];
    float* po = pO + (size_t)prow * 64;
    po[ 0 + mn] = o0[i];
    po[16 + mn] = o1[i];
    po[32 + mn] = o2[i];
    po[48 + mn] = o3[i];
  }
}

// ---------------- 5c: combine chunks (fixed order -> deterministic), scatter into ctx ----------------
__global__ void attn_combine_kernel(const int* __restrict__ topIdx,
                                    const float* __restrict__ psum, const float* __restrict__ pO,
                                    float* __restrict__ ctx) {
  int bh = blockIdx.x, tid = threadIdx.x;
  const int* ti = topIdx + bh * NQ_;
  for (int idx = tid; idx < 48 * 64; idx += 256) {
    int row = idx >> 6, f = idx & 63;
    if (row < NQ_) {
      float s = 0.f, o = 0.f;
      for (int c = 0; c < NCH_; ++c) {
        int prow = (bh * NCH_ + c) * 48 + row;
        s += psum[prow];
        o += pO[(size_t)prow * 64 + f];
      }
      ctx[((size_t)bh * L_ + ti[row]) * F_ + f] = o / s;
    }
  }
}

// ---------------- 6: projection (8192x512) = ctx @ Wp^T + bp ----------------
// Block: 128 rows x 32 cols; each wave: 16 rows x 32 cols (2 N-tiles, A-fragment reuse)
__global__ void proj_kernel(const float* __restrict__ X, const float* __restrict__ Wp,
                            const float* __restrict__ bp, float* __restrict__ out) {
  __shared__ float Xs[128 * PAD_];   // 34.8 KB
  __shared__ float Bs[32 * PAD_];    // Bs[n][kk] = Wp[j0+n][kc+kk]
  int tid = threadIdx.x;
  int wave = tid >> 5, lane = tid & 31;
  int half = lane >> 4, mn = lane & 15, koff = half * 2;
  int rblk = blockIdx.x * 128;
  int j0 = blockIdx.y * 32;
  v8f c00 = {}, c01 = {}, c10 = {}, c11 = {};
  for (int kc = 0; kc < D_; kc += 64) {
    __syncthreads();
    for (int i = tid; i < 128; i += 256) {         // Wp panel: 32 rows x 4 float4
      int n = i >> 2, c4 = (i & 3) * 4;
      cp16(&Bs[n * PAD_ + c4], &Wp[(j0 + n) * D_ + kc + c4]);
    }
    for (int i = tid; i < 512; i += 256) {         // X tile: 128 rows x 4 float4
      int r = i >> 2, c4 = (i & 3) * 4;
      cp16(&Xs[r * PAD_ + c4], &X[(size_t)(rblk + r) * D_ + kc + c4]);
    }
    cp_wait();
    __syncthreads();
#pragma unroll
    for (int kk = 0; kk < 64; kk += 8) {
      v2f a, b0, b1;
      a.x = Xs[(wave * 16 + mn) * PAD_ + kk + koff];
      a.y = Xs[(wave * 16 + mn) * PAD_ + kk + koff + 1];
      b0.x = Bs[mn * PAD_ + kk + koff];
      b0.y = Bs[mn * PAD_ + kk + koff + 1];
      b1.x = Bs[(16 + mn) * PAD_ + kk + koff];
      b1.y = Bs[(16 + mn) * PAD_ + kk + koff + 1];
      c00 = wmma4(a, b0, c00);
      c10 = wmma4(a, b1, c10);
      a.x = Xs[(wave * 16 + mn) * PAD_ + kk + 4 + koff];
      a.y = Xs[(wave * 16 + mn) * PAD_ + kk + 4 + koff + 1];
      b0.x = Bs[mn * PAD_ + kk + 4 + koff];
      b0.y = Bs[mn * PAD_ + kk + 4 + koff + 1];
      b1.x = Bs[(16 + mn) * PAD_ + kk + 4 + koff];
      b1.y = Bs[(16 + mn) * PAD_ + kk + 4 + koff + 1];
      c01 = wmma4(a, b0, c01);
      c11 = wmma4(a, b1, c11);
    }
  }
  v8f c0 = c00 + c01;
  v8f c1 = c10 + c11;
#pragma unroll
  for (int i = 0; i < 8; ++i) {
    int row = rblk + wave * 16 + i + 8 * half;
    out[(size_t)row * D_ + j0 + mn]      = c0[i] + bp[j0 + mn];
    out[(size_t)row * D_ + j0 + 16 + mn] = c1[i] + bp[j0 + 16 + mn];
  }
}

extern "C" void kernel_launch(void* const* d_in, const int* in_sizes, int n_in,
                              void* d_out, int out_size, void* d_ws, size_t ws_size,
                              hipStream_t stream) {
  (void)in_sizes; (void)n_in; (void)out_size; (void)ws_size;
  const float* q   = (const float*)d_in[0];
  const float* k   = (const float*)d_in[1];
  const float* v   = (const float*)d_in[2];
  const int* key_idx = (const int*)d_in[3];
  const float* Wp  = (const float*)d_in[4];
  const float* bp  = (const float*)d_in[5];
  float* out = (float*)d_out;

  char* ws = (char*)d_ws;
  float* mbuf   = (float*)(ws + 0);         // 65536 floats
  int*   topIdx = (int*)  (ws + 262144);    // 1280 ints
  float* vmean  = (float*)(ws + 270336);    // 2048 floats
  float* pmax   = (float*)(ws + 278528);    // 32*8*48 floats
  float* psum   = (float*)(ws + 327680);    // 32*8*48 floats
  float* pO     = (float*)(ws + 376832);    // 32*8*48*64 floats (3 MB)
  float* ctx    = (float*)(ws + 3522560);   // 4M floats: proj input viewed (8192,512)

  vmean_kernel       <<<dim3(BH_),              dim3(256), 0, stream>>>(v, vmean);
  ctxfill_kernel     <<<dim3(BH_*L_*F_/1024),   dim3(256), 0, stream>>>(vmean, (float4*)ctx);
  sparsity_kernel    <<<dim3(BH_*L_/8),         dim3(256), 0, stream>>>(q, k, key_idx, mbuf);
  topk_kernel        <<<dim3(BH_),              dim3(256), 0, stream>>>(mbuf, topIdx);
  attn_max_kernel    <<<dim3(BH_, NCH_),        dim3(96),  0, stream>>>(q, k, topIdx, pmax);
  attn_pv_kernel     <<<dim3(BH_, NCH_),        dim3(96),  0, stream>>>(q, k, v, topIdx, pmax, psum, pO);
  attn_combine_kernel<<<dim3(BH_),              dim3(256), 0, stream>>>(topIdx, psum, pO, ctx);
  proj_kernel        <<<dim3((B_*L_)/128, D_/32), dim3(256), 0, stream>>>(ctx, Wp, bp, out);
}